// LengthRegulator_20899310862777
// MI455X (gfx1250) — compile-verified
//
#include <hip/hip_runtime.h>
#include <hip/hip_bf16.h>
#include <stdint.h>

// Problem constants (fixed in the reference): B=64, S=1024, D=512, T=2048.
constexpr int B = 64;
constexpr int S = 1024;
constexpr int D = 512;
constexpr int T = 2048;

typedef float v4f __attribute__((ext_vector_type(4)));

// ---------------------------------------------------------------------------
// Kernel 1: per-batch inclusive scan of durations -> cum (int32), in d_ws.
// One block of 1024 threads (32 wave32's) per batch row; Hillis-Steele in LDS.
// ---------------------------------------------------------------------------
__global__ __launch_bounds__(1024)
void lr_cumsum_kernel(const int* __restrict__ durations, int* __restrict__ cum)
{
    __shared__ int s[S];
    const int b   = blockIdx.x;
    const int tid = threadIdx.x;

    s[tid] = durations[(size_t)b * S + tid];
    __syncthreads();

    #pragma unroll
    for (int off = 1; off < S; off <<= 1) {
        const int v   = s[tid];
        const int add = (tid >= off) ? s[tid - off] : 0;
        __syncthreads();
        s[tid] = v + add;
        __syncthreads();
    }

    cum[(size_t)b * S + tid] = s[tid];
}

// ---------------------------------------------------------------------------
// Kernel 2: gather + mask. One block per (batch, tile of TT frames).
// cum row is staged into LDS with CDNA5 async global->LDS loads (ASYNCcnt),
// searches are wave-uniform LDS broadcasts, output rows stream out with
// non-temporal b128 stores (keep x resident in the 192MB L2, don't pollute
// it with the 256MB write-once output).
// ---------------------------------------------------------------------------
constexpr int TT      = 64;    // frames per block
constexpr int THREADS = 256;   // 8 wave32's
constexpr int WAVES   = THREADS / 32;

__global__ __launch_bounds__(THREADS)
void lr_gather_kernel(const float* __restrict__ x,
                      const int*   __restrict__ cum,
                      float*       __restrict__ out,
                      float*       __restrict__ mask_out)
{
    __shared__ int scum[S];

    const int tilesPerB = T / TT;              // 32
    const int b    = blockIdx.x / tilesPerB;
    const int tile = blockIdx.x % tilesPerB;
    const int tid  = threadIdx.x;

    // ---- async-load this batch's cum row (4KB) into LDS ----
    {
        const int* row = cum + (size_t)b * S;
        // Flat LDS addresses carry the wave-relative LDS byte offset in
        // addr[31:0] (ISA 10.2) — truncate to get
        // the LDS offset the async instruction's VDST expects.
        const unsigned lds_base = (unsigned)(uintptr_t)&scum[0];
        #pragma unroll
        for (int k = 0; k < S / THREADS; ++k) {
            const unsigned byteoff = (unsigned)(tid + k * THREADS) * 4u;
            const unsigned ldsoff  = lds_base + byteoff;
            asm volatile("global_load_async_to_lds_b32 %0, %1, %2"
                         :
                         : "v"(ldsoff), "v"(byteoff), "s"(row)
                         : "memory");
        }
        asm volatile("s_wait_asynccnt 0" ::: "memory");
    }
    __syncthreads();

    const int wave = tid >> 5;
    const int lane = tid & 31;

    const int length = scum[S - 1];
    const size_t xrow_base = (size_t)b * S * D;
    const size_t orow_base = (size_t)b * T * D;

    for (int tl = wave; tl < TT; tl += WAVES) {
        const int  t    = tile * TT + tl;      // uniform across the wave
        const bool dead = (t >= length);

        // searchsorted(cum, t, side='right'): first i with cum[i] > t
        int lo = 0, hi = S;
        while (lo < hi) {
            const int mid = (lo + hi) >> 1;
            if (scum[mid] <= t) lo = mid + 1; else hi = mid;
        }
        const int idx = (lo < S - 1) ? lo : (S - 1);

        v4f* dst = (v4f*)(out + orow_base + (size_t)t * D);
        if (dead) {
            const v4f z = {0.f, 0.f, 0.f, 0.f};
            #pragma unroll
            for (int i = lane; i < D / 4; i += 32)
                __builtin_nontemporal_store(z, dst + i);
        } else {
            const v4f* src = (const v4f*)(x + xrow_base + (size_t)idx * D);
            #pragma unroll
            for (int i = lane; i < D / 4; i += 32) {
                const v4f v = src[i];
                __builtin_nontemporal_store(v, dst + i);
            }
        }

        if (lane == 0)
            mask_out[(size_t)b * T + t] = dead ? 1.0f : 0.0f;
    }
}

// ---------------------------------------------------------------------------
extern "C" void kernel_launch(void* const* d_in, const int* in_sizes, int n_in,
                              void* d_out, int out_size, void* d_ws, size_t ws_size,
                              hipStream_t stream)
{
    const float* x         = (const float*)d_in[0];
    const int*   durations = (const int*)d_in[1];
    // d_in[2] is max_length (scalar, statically 2048)

    int*   cum      = (int*)d_ws;                       // B*S int32 = 256 KB
    float* out      = (float*)d_out;                    // B*T*D f32
    float* mask_out = out + (size_t)B * T * D;          // B*T (0.0 / 1.0)

    lr_cumsum_kernel<<<B, S, 0, stream>>>(durations, cum);
    lr_gather_kernel<<<B * (T / TT), THREADS, 0, stream>>>(x, cum, out, mask_out);
}